// PhysicsResidual_15144054686218
// MI455X (gfx1250) — compile-verified
//
#include <hip/hip_runtime.h>
#include <cmath>

typedef float v2f __attribute__((ext_vector_type(2)));
typedef float v8f __attribute__((ext_vector_type(8)));

#define WIDTH  128
#define NHID   5        // hidden-to-hidden layers
#define JETS   9        // 1 value + 4 first-deriv + 4 second-deriv
#define TPB    256      // 8 waves of 32
#define PTS    16       // points per workgroup (WMMA N)
#define KK     64       // WIDTH/2 k-pair rows
#define ROWLEN 34       // 16 points * 2 floats + 2 pad floats (bank stagger)

__device__ __forceinline__ int hidx(int j, int kk, int n2) {
  return (j * KK + kk) * ROWLEN + n2;
}

__global__ __launch_bounds__(TPB)
void wavepinn_resid_kernel(
    const float* __restrict__ t,  const float* __restrict__ x,
    const float* __restrict__ y,  const float* __restrict__ zc,
    const float* __restrict__ uW_in,  const float* __restrict__ ub_in,
    const float* __restrict__ uW_hid, const float* __restrict__ ub_hid,
    const float* __restrict__ uW_out, const float* __restrict__ ub_out,
    const float* __restrict__ pW_in,  const float* __restrict__ pb_in,
    const float* __restrict__ pW_hid, const float* __restrict__ pb_hid,
    const float* __restrict__ pW_out, const float* __restrict__ pb_out,
    float* __restrict__ out, int N)
{
  extern __shared__ float hbuf[];   // [JETS][KK][ROWLEN] : jets, k-pair rows, 2*point

  const int tid  = threadIdx.x;
  const int lane = tid & 31;
  const int wave = tid >> 5;        // 0..7 -> M-tile
  const int n    = lane & 15;       // point column 0..15
  const int hi   = lane >> 4;       // row/k half select
  const int m0   = wave * 16;
  const int p0   = blockIdx.x * PTS;

  // ---------------- input layer 4 -> 128 : jets computed directly ----------------
  {
    const int p = p0 + n;
    const float pt0 = t[p], pt1 = x[p], pt2 = y[p], pt3 = zc[p];
    #pragma unroll
    for (int rp = 0; rp < 4; ++rp) {
      v2f v[JETS];
      #pragma unroll
      for (int s = 0; s < 2; ++s) {
        const int m = m0 + 8 * hi + 2 * rp + s;
        const float w0 = uW_in[m*4+0], w1 = uW_in[m*4+1];
        const float w2 = uW_in[m*4+2], w3 = uW_in[m*4+3];
        const float zv = fmaf(w0, pt0, fmaf(w1, pt1, fmaf(w2, pt2, fmaf(w3, pt3, ub_in[m]))));
        const float a  = tanhf(zv);
        const float g  = 1.0f - a * a;     // tanh'
        const float h2 = -2.0f * a * g;    // tanh''
        v[0][s] = a;
        v[1][s] = g * w0;       v[2][s] = g * w1;
        v[3][s] = g * w2;       v[4][s] = g * w3;
        v[5][s] = h2 * w0 * w0; v[6][s] = h2 * w1 * w1;
        v[7][s] = h2 * w2 * w2; v[8][s] = h2 * w3 * w3;
      }
      const int kk = (m0 >> 1) + 4 * hi + rp;
      #pragma unroll
      for (int j = 0; j < JETS; ++j)
        *reinterpret_cast<v2f*>(&hbuf[hidx(j, kk, 2 * n)]) = v[j];
    }
  }
  __syncthreads();

  // ---------------- 5 hidden layers: jets GEMM on fp32 WMMA ----------------
  const int arow = m0 + n;                // this lane's A row
  for (int layer = 0; layer < NHID; ++layer) {
    const float* W = uW_hid + layer * WIDTH * WIDTH;
    const float* b = ub_hid + layer * WIDTH;

    // warm next layer's A slice into cache while we compute this one
    if (layer + 1 < NHID) {
      const float* Wn = W + WIDTH * WIDTH;
      #pragma unroll
      for (int c = 0; c < 4; ++c)
        __builtin_prefetch(&Wn[(m0 + n) * WIDTH + 32 * c], 0, 3);
    }

    v8f acc[JETS] = {};                   // 9 x 16x16 f32 accumulators (72 VGPRs)

    const float* aptr = W + arow * WIDTH + 2 * hi;  // A frag: W[arow][4*step + 2*hi .. +1]
    const float* bptr[JETS];              // per-jet LDS base, step stride = 2*ROWLEN floats
    #pragma unroll
    for (int j = 0; j < JETS; ++j) bptr[j] = &hbuf[hidx(j, hi, 2 * n)];

    // software pipeline: A two steps ahead, B one step ahead
    v2f acur, anxt, bcur[JETS], bnxt[JETS];
    acur = *reinterpret_cast<const v2f*>(aptr);
    anxt = *reinterpret_cast<const v2f*>(aptr + 4);
    #pragma unroll
    for (int j = 0; j < JETS; ++j)
      bcur[j] = *reinterpret_cast<const v2f*>(bptr[j]);

    #pragma unroll
    for (int step = 0; step < 32; ++step) {
      const int pn = (step + 1) & 31;     // wrap: tail prefetches are harmless re-reads
      #pragma unroll
      for (int j = 0; j < JETS; ++j)
        bnxt[j] = *reinterpret_cast<const v2f*>(bptr[j] + pn * 2 * ROWLEN);
      const v2f a_use = acur;
      acur = anxt;
      anxt = *reinterpret_cast<const v2f*>(aptr + 4 * ((step + 2) & 31));
      #pragma unroll
      for (int j = 0; j < JETS; ++j)
        acc[j] = __builtin_amdgcn_wmma_f32_16x16x4_f32(
            false, a_use, false, bcur[j], (short)0, acc[j], false, false);
      #pragma unroll
      for (int j = 0; j < JETS; ++j) bcur[j] = bnxt[j];   // renamed away by unroll
    }
    __syncthreads();                      // all LDS reads of this layer done

    // tanh jet chain rule, write activations back in place
    #pragma unroll
    for (int rp = 0; rp < 4; ++rp) {
      v2f v[JETS];
      #pragma unroll
      for (int s = 0; s < 2; ++s) {
        const int r = 2 * rp + s;
        const int m = m0 + 8 * hi + r;
        const float zv = acc[0][r] + b[m];
        const float a  = tanhf(zv);
        const float g  = 1.0f - a * a;
        const float h2 = -2.0f * a * g;
        v[0][s] = a;
        #pragma unroll
        for (int i = 0; i < 4; ++i) {
          const float d1 = acc[1 + i][r];
          v[1 + i][s] = g * d1;
          v[5 + i][s] = fmaf(g, acc[5 + i][r], h2 * d1 * d1);
        }
      }
      const int kk = (m0 >> 1) + 4 * hi + rp;
      #pragma unroll
      for (int j = 0; j < JETS; ++j)
        *reinterpret_cast<v2f*>(&hbuf[hidx(j, kk, 2 * n)]) = v[j];
    }
    __syncthreads();
  }

  // ---------------- output layer + psi network + residual ----------------
  if (tid < PTS) {
    const int p = p0 + tid;
    float d[4] = {0.f, 0.f, 0.f, 0.f};  // u_tt, u_xx, u_yy, u_zz
    for (int kk = 0; kk < KK; ++kk) {
      const float w0 = uW_out[2 * kk], w1 = uW_out[2 * kk + 1];
      #pragma unroll
      for (int k = 0; k < 4; ++k) {
        const v2f s2 = *reinterpret_cast<const v2f*>(&hbuf[hidx(5 + k, kk, 2 * tid)]);
        d[k] = fmaf(w0, s2.x, fmaf(w1, s2.y, d[k]));
      }
    }
    // psi MLP 4 -> 32 -> 32 -> 1 (tiny; serial per point)
    const float pt0 = t[p], pt1 = x[p], pt2 = y[p], pt3 = zc[p];
    float h1[32], h2a[32];
    for (int i = 0; i < 32; ++i) {
      float zv = fmaf(pW_in[i*4+0], pt0, fmaf(pW_in[i*4+1], pt1,
                 fmaf(pW_in[i*4+2], pt2, fmaf(pW_in[i*4+3], pt3, pb_in[i]))));
      h1[i] = tanhf(zv);
    }
    for (int i = 0; i < 32; ++i) {
      float zv = pb_hid[i];
      for (int j = 0; j < 32; ++j) zv = fmaf(pW_hid[i*32+j], h1[j], zv);
      h2a[i] = tanhf(zv);
    }
    float psi = pb_out[0];
    for (int i = 0; i < 32; ++i) psi = fmaf(pW_out[i], h2a[i], psi);

    const float c   = 1.0f + psi;       // C0 == 1
    const float lap = d[1] + d[2] + d[3];
    float resid = d[0] - c * c * lap;
    if (__builtin_isnan(d[0]) || __builtin_isnan(d[1]) || __builtin_isnan(d[2]) ||
        __builtin_isnan(d[3]) || __builtin_isnan(c))
      resid = __builtin_nanf("");
    out[p] = resid;
  }
  (void)ub_out; // derivative of linear output bias is zero
}

extern "C" void kernel_launch(void* const* d_in, const int* in_sizes, int n_in,
                              void* d_out, int out_size, void* d_ws, size_t ws_size,
                              hipStream_t stream) {
  const float* t      = (const float*)d_in[0];
  const float* x      = (const float*)d_in[1];
  const float* y      = (const float*)d_in[2];
  const float* z      = (const float*)d_in[3];
  const float* uW_in  = (const float*)d_in[4];
  const float* ub_in  = (const float*)d_in[5];
  const float* uW_hid = (const float*)d_in[6];
  const float* ub_hid = (const float*)d_in[7];
  const float* uW_out = (const float*)d_in[8];
  const float* ub_out = (const float*)d_in[9];
  const float* pW_in  = (const float*)d_in[10];
  const float* pb_in  = (const float*)d_in[11];
  const float* pW_hid = (const float*)d_in[12];
  const float* pb_hid = (const float*)d_in[13];
  const float* pW_out = (const float*)d_in[14];
  const float* pb_out = (const float*)d_in[15];

  const int N = in_sizes[0];
  const int grid = N / PTS;
  const size_t shmem = (size_t)JETS * KK * ROWLEN * sizeof(float); // 78,336 B

  hipFuncSetAttribute(reinterpret_cast<const void*>(wavepinn_resid_kernel),
                      hipFuncAttributeMaxDynamicSharedMemorySize, (int)shmem);

  wavepinn_resid_kernel<<<grid, TPB, shmem, stream>>>(
      t, x, y, z, uW_in, ub_in, uW_hid, ub_hid, uW_out, ub_out,
      pW_in, pb_in, pW_hid, pb_hid, pW_out, pb_out,
      (float*)d_out, N);
}